// Linear_act_sp_4690104287268
// MI455X (gfx1250) — compile-verified
//
#include <hip/hip_runtime.h>

typedef __bf16 bf16_t;
typedef __attribute__((ext_vector_type(4)))  float  v4f;
typedef __attribute__((ext_vector_type(8)))  float  v8f;
typedef __attribute__((ext_vector_type(8)))  __bf16 v8bf;
typedef __attribute__((ext_vector_type(16))) __bf16 v16bf;
typedef __attribute__((ext_vector_type(32))) __bf16 v32bf;

// Async global->LDS copy (ASYNCcnt-tracked) and its wait, via inline asm
// (no probe-confirmed builtins for these; asm is portable per CDNA5_HIP.md).
__device__ __forceinline__ void async_ld_b128(unsigned lds_off, const void* g) {
  asm volatile("global_load_async_to_lds_b128 %0, %1, off"
               :: "v"(lds_off), "v"(g) : "memory");
}
__device__ __forceinline__ void wait_async0() {
  asm volatile("s_wait_asynccnt 0x0" ::: "memory");
}

// ---------------------------------------------------------------------------
// Kernel 1: 2:4 magnitude prune + pack activations to bf16 (SWMMAC A layout).
// One thread per (row, 32-K chunk): 8 groups of 4 -> 16 packed bf16 values +
// one u32 of 2-bit index pairs (group g at bits [4g+3:4g], lo<hi).
// ---------------------------------------------------------------------------
__global__ __launch_bounds__(256) void prune_pack_kernel(
    const float* __restrict__ x, bf16_t* __restrict__ ap,
    unsigned* __restrict__ meta, long total_chunks) {
  long tid = (long)blockIdx.x * blockDim.x + threadIdx.x;
  if (tid >= total_chunks) return;
  const int K = 4096, CH = K / 32;
  long row = tid / CH;
  int  c   = (int)(tid - row * CH);

  const float* src = x + row * (long)K + (long)c * 32;
  float v[32];
#pragma unroll
  for (int i = 0; i < 8; ++i) {
    v4f q = *(const v4f*)(src + i * 4);
    v[i * 4 + 0] = q[0]; v[i * 4 + 1] = q[1];
    v[i * 4 + 2] = q[2]; v[i * 4 + 3] = q[3];
  }

  unsigned idxword = 0;
  union { v16bf vec; bf16_t e[16]; } pk;
#pragma unroll
  for (int g = 0; g < 8; ++g) {
    float a[4], m[4];
#pragma unroll
    for (int j = 0; j < 4; ++j) { a[j] = v[g * 4 + j]; m[j] = fabsf(a[j]); }
    int i0 = 0;
#pragma unroll
    for (int j = 1; j < 4; ++j) if (m[j] > m[i0]) i0 = j;
    int i1 = (i0 == 0) ? 1 : 0;
#pragma unroll
    for (int j = 0; j < 4; ++j)
      if (j != i0 && j != i1 && m[j] > m[i1]) i1 = j;
    int lo = i0 < i1 ? i0 : i1;
    int hi = i0 < i1 ? i1 : i0;
    pk.e[g * 2 + 0] = (bf16_t)a[lo];
    pk.e[g * 2 + 1] = (bf16_t)a[hi];
    idxword |= (unsigned)(lo | (hi << 2)) << (g * 4);
  }
  *(v16bf*)(ap + row * (long)(K / 2) + (long)c * 16) = pk.vec;
  meta[row * (long)CH + c] = idxword;
}

// ---------------------------------------------------------------------------
// Kernel 2: weight fp32 -> bf16 ([N, K] row-major preserved; RNE).
// ---------------------------------------------------------------------------
__global__ __launch_bounds__(256) void wconv_kernel(
    const float* __restrict__ w, bf16_t* __restrict__ wb, long n8) {
  long i = (long)blockIdx.x * blockDim.x + threadIdx.x;
  if (i >= n8) return;
  const float* s = w + i * 8;
  v4f q0 = *(const v4f*)(s);
  v4f q1 = *(const v4f*)(s + 4);
  union { v8bf vec; bf16_t e[8]; } o;
#pragma unroll
  for (int j = 0; j < 4; ++j) { o.e[j] = (bf16_t)q0[j]; o.e[4 + j] = (bf16_t)q1[j]; }
  *(v8bf*)(wb + i * 8) = o.vec;
}

// ---------------------------------------------------------------------------
// Kernel 3: 2:4-sparse GEMM via V_SWMMAC_F32_16X16X64_BF16 with async-to-LDS
// double-buffered K-slab staging.
//   Block = 8 waves -> 128(M) x 128(N); each wave: 2x4 tiles of 16x16.
//   Per K=64 step the block stages: packed A slab 128x32 bf16 (8 KB) and
//   B slab 128x64 bf16 (16 KB); 2 buffers = 48 KB LDS.
// Iteration protocol (one barrier, full overlap):
//   s_wait_asynccnt 0  -> own group-k loads landed
//   __syncthreads()    -> all waves' writes visible; all waves also finished
//                         reading buf[k-1] (their compute(k-1) precedes this)
//   issue group k+1    -> into the other buffer (safe by the barrier above)
//   compute group k    -> ds_load fragments + swmmac, overlapping k+1 loads
// ---------------------------------------------------------------------------
__global__ __launch_bounds__(256) void spgemm_kernel(
    const bf16_t* __restrict__ ap, const unsigned* __restrict__ meta,
    const bf16_t* __restrict__ wb, float* __restrict__ out) {
  const int K = 4096, N = 4096;
  const int KP = K / 2;     // packed row length (bf16 elements)
  const int CH = K / 32;    // meta words per row
  constexpr int ABYTES  = 128 * 64;          // 8 KB
  constexpr int BBYTES  = 128 * 128;         // 16 KB
  constexpr int BUFBYTES = ABYTES + BBYTES;  // 24 KB

  __shared__ __align__(16) char smem[2 * BUFBYTES];

  const int tid  = threadIdx.x;
  const int lane = tid & 31;
  const int wv   = tid >> 5;
  const int wm   = wv & 3;   // 4 waves along M
  const int wn   = wv >> 2;  // 2 waves along N
  const int lr   = lane & 15;
  const int lh   = lane >> 4;

  const long mblk = (long)blockIdx.x * 128;
  const long nblk = (long)blockIdx.y * 128;

  // LDS byte offset of smem (generic LDS address keeps offset in addr[31:0])
  const unsigned ldsbase = (unsigned)(size_t)&smem[0];

  auto issue_slab = [&](int k64, int buf) {
    const unsigned ab = ldsbase + (unsigned)(buf * BUFBYTES);
    const unsigned bb = ab + ABYTES;
#pragma unroll
    for (int i = 0; i < 2; ++i) {      // A: 512 x 16B chunks, 2 per thread
      int c = tid + i * 256;
      int row = c >> 2, q = c & 3;
      async_ld_b128(ab + (unsigned)(row * 64 + q * 16),
                    ap + (mblk + row) * (long)KP + k64 * 32 + q * 8);
    }
#pragma unroll
    for (int i = 0; i < 4; ++i) {      // B: 1024 x 16B chunks, 4 per thread
      int c = tid + i * 256;
      int row = c >> 3, q = c & 7;
      async_ld_b128(bb + (unsigned)(row * 128 + q * 16),
                    wb + (nblk + row) * (long)K + k64 * 64 + q * 8);
    }
  };

  v8f acc[2][4];
#pragma unroll
  for (int mt = 0; mt < 2; ++mt)
#pragma unroll
    for (int nt = 0; nt < 4; ++nt) acc[mt][nt] = 0.f;

  const unsigned* mRow[2];
#pragma unroll
  for (int mt = 0; mt < 2; ++mt)
    mRow[mt] = meta + (mblk + wm * 32 + mt * 16 + lr) * (long)CH;

  issue_slab(0, 0);

  const int NK = K / 64;  // 64 steps
  for (int k64 = 0; k64 < NK; ++k64) {
    const int buf = k64 & 1;
    wait_async0();
    __syncthreads();
    if (k64 + 1 < NK) {
      issue_slab(k64 + 1, buf ^ 1);
      __builtin_prefetch(mRow[0] + (k64 + 1) * 2, 0, 1);  // meta prefetch
    }

    const char* sm = smem + buf * BUFBYTES;

    // A fragments (16-bit 16x32 packed layout) + sparse index word
    union { v16bf vec; v8bf h[2]; } afr[2];
    int idxw[2];
#pragma unroll
    for (int mt = 0; mt < 2; ++mt) {
      const char* ab = sm + (wm * 32 + mt * 16 + lr) * 64 + lh * 16;
      afr[mt].h[0] = *(const v8bf*)(ab);        // packed-K {0..7}/{8..15}
      afr[mt].h[1] = *(const v8bf*)(ab + 32);   // packed-K {16..23}/{24..31}
      idxw[mt] = (int)mRow[mt][k64 * 2 + lh];
    }
    // B fragments (64x16 bf16 layout)
    union { v32bf vec; v8bf h[4]; } bfr[4];
#pragma unroll
    for (int nt = 0; nt < 4; ++nt) {
      const char* bp = sm + ABYTES + (wn * 64 + nt * 16 + lr) * 128 + lh * 32;
      bfr[nt].h[0] = *(const v8bf*)(bp);        // K = half*16 + 0..7
      bfr[nt].h[1] = *(const v8bf*)(bp + 16);   // K = half*16 + 8..15
      bfr[nt].h[2] = *(const v8bf*)(bp + 64);   // K = 32 + half*16 + 0..7
      bfr[nt].h[3] = *(const v8bf*)(bp + 80);   // K = 32 + half*16 + 8..15
    }

#pragma unroll
    for (int mt = 0; mt < 2; ++mt)
#pragma unroll
      for (int nt = 0; nt < 4; ++nt)
        acc[mt][nt] = __builtin_amdgcn_swmmac_f32_16x16x64_bf16(
            false, afr[mt].vec, false, bfr[nt].vec,
            acc[mt][nt], idxw[mt], false, false);
  }

  // Epilogue: C/D 16x16 f32 layout — VGPR r: lanes 0-15 -> M=r, lanes 16-31 -> M=r+8
#pragma unroll
  for (int mt = 0; mt < 2; ++mt) {
#pragma unroll
    for (int nt = 0; nt < 4; ++nt) {
      long nc = nblk + wn * 64 + nt * 16 + lr;
#pragma unroll
      for (int r = 0; r < 8; ++r) {
        long mr = mblk + wm * 32 + mt * 16 + lh * 8 + r;
        out[mr * (long)N + nc] = acc[mt][nt][r];
      }
    }
  }
}

// ---------------------------------------------------------------------------
extern "C" void kernel_launch(void* const* d_in, const int* in_sizes, int n_in,
                              void* d_out, int out_size, void* d_ws, size_t ws_size,
                              hipStream_t stream) {
  const float* x = (const float*)d_in[0];    // [8, 2048, 4096] fp32
  const float* w = (const float*)d_in[1];    // [4096, 4096] fp32
  float* out = (float*)d_out;                // [8, 2048, 4096] fp32

  const long M = 8L * 2048L;  // 16384 rows
  const long K = 4096, N = 4096;

  // Workspace: [packed A bf16 64MB][meta u32 8MB][weight bf16 32MB]
  char* ws = (char*)d_ws;
  bf16_t* apk = (bf16_t*)ws;
  size_t apBytes = (size_t)M * (K / 2) * sizeof(bf16_t);
  unsigned* meta = (unsigned*)(ws + apBytes);
  size_t metaBytes = (size_t)M * (K / 32) * sizeof(unsigned);
  bf16_t* wbf = (bf16_t*)(ws + apBytes + metaBytes);

  long chunks = M * (K / 32);
  prune_pack_kernel<<<dim3((unsigned)((chunks + 255) / 256)), dim3(256), 0, stream>>>(
      x, apk, meta, chunks);

  long n8 = N * K / 8;
  wconv_kernel<<<dim3((unsigned)((n8 + 255) / 256)), dim3(256), 0, stream>>>(w, wbf, n8);

  dim3 grid((unsigned)(M / 128), (unsigned)(N / 128));  // 128 x 32 blocks
  spgemm_kernel<<<grid, dim3(256), 0, stream>>>(apk, meta, wbf, out);
}